// TEGNNLayer_74113955660244
// MI455X (gfx1250) — compile-verified
//
#include <hip/hip_runtime.h>
#include <hip/hip_bf16.h>

typedef __attribute__((ext_vector_type(2))) float v2f;
typedef __attribute__((ext_vector_type(8))) float v8f;

#define WMMA_F32X4(a, b, c) \
    __builtin_amdgcn_wmma_f32_16x16x4_f32(false, (a), false, (b), (short)0, (c), false, false)

#define D_DIM 128
#define K_PKT 8
#define B_GRAPHS 64
#define LDS_STRIDE 132   // 128 + 4 pad: 16 rows hit 16 distinct banks on column walks

// ---------------------------------------------------------------------------
// Zero workspace regions (agg, pocket_sum, pocket_count)
// ---------------------------------------------------------------------------
__global__ void zero_kernel(float* __restrict__ p, long n) {
    long i = (long)blockIdx.x * blockDim.x + threadIdx.x;
    long stride = (long)gridDim.x * blockDim.x;
    for (; i < n; i += stride) p[i] = 0.0f;
}

// ---------------------------------------------------------------------------
// Edge kernel: emb = edge_attr @ We (+be); msg = relu(x[src]+emb); atomic add
// at agg[dst]. One wave32 handles 16 edges; 8 column tiles x 4 K-steps of
// v_wmma_f32_16x16x4_f32 per group.
// ---------------------------------------------------------------------------
__global__ __launch_bounds__(256) void edge_aggregate_kernel(
    const float* __restrict__ x, const int* __restrict__ ei,
    const float* __restrict__ ea, const float* __restrict__ We,
    const float* __restrict__ be, float* __restrict__ agg, int E)
{
    const int lane = threadIdx.x & 31;
    const int wave = threadIdx.x >> 5;
    const int group = blockIdx.x * 8 + wave;   // 16 edges per group
    const int nGroups = E >> 4;                // E divisible by 16
    if (group >= nGroups) return;              // wave-uniform exit
    const int e0  = group << 4;
    const int row = lane & 15;
    const int hi  = lane >> 4;                 // 0 or 1 (half-wave)

    // Edge ids for the 8 C-matrix rows this lane owns (M = i + hi*8)
    int srcs[8], dsts[8];
    #pragma unroll
    for (int i = 0; i < 8; ++i) {
        int m = i + hi * 8;
        srcs[i] = ei[e0 + m];
        dsts[i] = ei[E + e0 + m];
    }

    // A operand (16x4 tiles of edge_attr), invariant across column tiles.
    // Lane layout: M = row, K = k0 + 2*hi (+1 in .y)
    v2f aop[4];
    #pragma unroll
    for (int ks = 0; ks < 4; ++ks) {
        int kk = ks * 4 + hi * 2;
        aop[ks].x = ea[(e0 + row) * 16 + kk];
        aop[ks].y = ea[(e0 + row) * 16 + kk + 1];
    }

    #pragma unroll 1
    for (int n0 = 0; n0 < D_DIM; n0 += 16) {
        v8f c = {0.f, 0.f, 0.f, 0.f, 0.f, 0.f, 0.f, 0.f};
        #pragma unroll
        for (int ks = 0; ks < 4; ++ks) {
            int kk = ks * 4 + hi * 2;
            v2f b;  // B layout: N = row, K = kk (+1 in .y)
            b.x = We[kk * D_DIM + n0 + row];
            b.y = We[(kk + 1) * D_DIM + n0 + row];
            c = WMMA_F32X4(aop[ks], b, c);
        }
        const int n = n0 + row;
        const float ben = be[n];
        #pragma unroll
        for (int i = 0; i < 8; ++i) {
            float msg = x[srcs[i] * D_DIM + n] + c[i] + ben;
            msg = msg > 0.f ? msg : 0.f;
            atomicAdd(&agg[dsts[i] * D_DIM + n], msg);
        }
    }
}

// ---------------------------------------------------------------------------
// Node MLP: h = (1+eps)*x + agg ; x_res = relu(h@W1+b1)@W2+b2
// Wave per 16-node tile; h and hidden staged in padded LDS; 512 WMMAs/tile.
// ---------------------------------------------------------------------------
__global__ __launch_bounds__(64) void node_mlp_kernel(
    const float* __restrict__ x, const float* __restrict__ agg,
    const float* __restrict__ W1, const float* __restrict__ b1,
    const float* __restrict__ W2, const float* __restrict__ b2,
    const float* __restrict__ epsp, float* __restrict__ xres, int N)
{
    __shared__ float lds_h[2][16 * LDS_STRIDE];
    __shared__ float lds_hid[2][16 * LDS_STRIDE];
    const int lane = threadIdx.x & 31;
    const int wave = threadIdx.x >> 5;
    const int tile = blockIdx.x * 2 + wave;
    const int nTiles = N >> 4;                 // N divisible by 16
    const int row = lane & 15;
    const int hi  = lane >> 4;
    const bool active = (tile < nTiles);
    const int base = tile << 4;
    float* hs = lds_h[wave];
    float* hd = lds_hid[wave];

    if (active) {
        const float opeps = 1.0f + epsp[0];
        for (int idx = lane; idx < 16 * D_DIM; idx += 32) {
            int m = idx >> 7, col = idx & 127;
            int g = (base + m) * D_DIM + col;
            hs[m * LDS_STRIDE + col] = opeps * x[g] + agg[g];
        }
    }
    __syncthreads();

    if (active) {
        // GEMM1: hidden = relu(h @ W1 + b1)
        #pragma unroll 1
        for (int n0 = 0; n0 < D_DIM; n0 += 16) {
            v8f c = {0.f, 0.f, 0.f, 0.f, 0.f, 0.f, 0.f, 0.f};
            #pragma unroll 4
            for (int k0 = 0; k0 < D_DIM; k0 += 4) {
                int kk = k0 + hi * 2;
                v2f a; a.x = hs[row * LDS_STRIDE + kk];
                       a.y = hs[row * LDS_STRIDE + kk + 1];
                v2f b; b.x = W1[kk * D_DIM + n0 + row];
                       b.y = W1[(kk + 1) * D_DIM + n0 + row];
                c = WMMA_F32X4(a, b, c);
            }
            int n = n0 + row;
            float bn = b1[n];
            #pragma unroll
            for (int i = 0; i < 8; ++i) {
                int m = i + hi * 8;
                float v = c[i] + bn;
                hd[m * LDS_STRIDE + n] = v > 0.f ? v : 0.f;
            }
        }
    }
    __syncthreads();

    if (active) {
        // GEMM2: x_res = hidden @ W2 + b2
        #pragma unroll 1
        for (int n0 = 0; n0 < D_DIM; n0 += 16) {
            v8f c = {0.f, 0.f, 0.f, 0.f, 0.f, 0.f, 0.f, 0.f};
            #pragma unroll 4
            for (int k0 = 0; k0 < D_DIM; k0 += 4) {
                int kk = k0 + hi * 2;
                v2f a; a.x = hd[row * LDS_STRIDE + kk];
                       a.y = hd[row * LDS_STRIDE + kk + 1];
                v2f b; b.x = W2[kk * D_DIM + n0 + row];
                       b.y = W2[(kk + 1) * D_DIM + n0 + row];
                c = WMMA_F32X4(a, b, c);
            }
            int n = n0 + row;
            float bn = b2[n];
            #pragma unroll
            for (int i = 0; i < 8; ++i) {
                int m = i + hi * 8;
                xres[(base + m) * D_DIM + n] = c[i] + bn;
            }
        }
    }
}

// ---------------------------------------------------------------------------
// Pocket pooling: psum[b,k,:] += x_res[n,:] * mask[n,k]; pcount[b,k] += mask
// One wave per (node, pocket) pair.
// ---------------------------------------------------------------------------
__global__ __launch_bounds__(256) void pocket_pool_kernel(
    const float* __restrict__ xres, const float* __restrict__ mask,
    const int* __restrict__ batch, float* __restrict__ psum,
    float* __restrict__ pcount, int N)
{
    const int lane = threadIdx.x & 31;
    const int wave = threadIdx.x >> 5;
    const long pair = (long)blockIdx.x * 8 + wave;
    if (pair >= (long)N * K_PKT) return;
    const int n = (int)(pair >> 3);
    const int k = (int)(pair & 7);
    const float mk = mask[n * K_PKT + k];
    if (mk == 0.f) return;                  // wave-uniform (same pair per wave)
    const int b = batch[n];
    const int o = (b * K_PKT + k) * D_DIM;
    if (lane == 0) atomicAdd(&pcount[b * K_PKT + k], mk);
    #pragma unroll
    for (int j = 0; j < 4; ++j) {
        int col = lane + 32 * j;
        atomicAdd(&psum[o + col], xres[n * D_DIM + col] * mk);
    }
}

// ---------------------------------------------------------------------------
// Pocket embedding GEMM: pemb = (psum / (count+1e-9)) @ Wv + bv  (512x128x128)
// Wave per 16-row tile, divide-by-count fused into A loads.
// ---------------------------------------------------------------------------
__global__ __launch_bounds__(64) void pocket_emb_kernel(
    const float* __restrict__ psum, const float* __restrict__ pcount,
    const float* __restrict__ Wv, const float* __restrict__ bv,
    float* __restrict__ pemb)
{
    const int lane = threadIdx.x & 31;
    const int wave = threadIdx.x >> 5;
    const int tile = blockIdx.x * 2 + wave;     // 32 tiles over 512 rows
    const int base = tile << 4;
    const int row = lane & 15;
    const int hi  = lane >> 4;
    const float invc = 1.0f / (pcount[base + row] + 1e-9f);

    #pragma unroll 1
    for (int n0 = 0; n0 < D_DIM; n0 += 16) {
        v8f c = {0.f, 0.f, 0.f, 0.f, 0.f, 0.f, 0.f, 0.f};
        #pragma unroll 4
        for (int k0 = 0; k0 < D_DIM; k0 += 4) {
            int kk = k0 + hi * 2;
            v2f a; a.x = psum[(base + row) * D_DIM + kk] * invc;
                   a.y = psum[(base + row) * D_DIM + kk + 1] * invc;
            v2f b; b.x = Wv[kk * D_DIM + n0 + row];
                   b.y = Wv[(kk + 1) * D_DIM + n0 + row];
            c = WMMA_F32X4(a, b, c);
        }
        int n = n0 + row;
        float bn = bv[n];
        #pragma unroll
        for (int i = 0; i < 8; ++i) {
            int m = i + hi * 8;
            pemb[(base + m) * D_DIM + n] = c[i] + bn;
        }
    }
}

// ---------------------------------------------------------------------------
// Finalize: x_res += masked pocket feedback; LayerNorm; ReLU. Wave per node.
// Reads & rewrites d_out in place (each lane owns its 4 columns).
// ---------------------------------------------------------------------------
__global__ __launch_bounds__(256) void finalize_kernel(
    float* __restrict__ out, const float* __restrict__ pemb,
    const float* __restrict__ mask, const int* __restrict__ batch,
    const float* __restrict__ gamma, const float* __restrict__ beta, int N)
{
    const int lane = threadIdx.x & 31;
    const int wave = threadIdx.x >> 5;
    const int n = blockIdx.x * 8 + wave;
    if (n >= N) return;
    const int b = batch[n];
    float v[4];
    #pragma unroll
    for (int j = 0; j < 4; ++j) v[j] = out[n * D_DIM + lane + 32 * j];
    #pragma unroll
    for (int k = 0; k < K_PKT; ++k) {
        float mk = mask[n * K_PKT + k];
        if (mk != 0.f) {                    // wave-uniform branch
            int o = (b * K_PKT + k) * D_DIM;
            #pragma unroll
            for (int j = 0; j < 4; ++j) v[j] += mk * pemb[o + lane + 32 * j];
        }
    }
    float s = v[0] + v[1] + v[2] + v[3];
    #pragma unroll
    for (int off = 16; off > 0; off >>= 1) s += __shfl_xor(s, off, 32);
    const float mu = s * (1.0f / 128.0f);
    float ss = 0.f;
    #pragma unroll
    for (int j = 0; j < 4; ++j) { float d = v[j] - mu; ss += d * d; }
    #pragma unroll
    for (int off = 16; off > 0; off >>= 1) ss += __shfl_xor(ss, off, 32);
    const float rstd = rsqrtf(ss * (1.0f / 128.0f) + 1e-5f);
    #pragma unroll
    for (int j = 0; j < 4; ++j) {
        int col = lane + 32 * j;
        float y = (v[j] - mu) * rstd * gamma[col] + beta[col];
        out[n * D_DIM + col] = y > 0.f ? y : 0.f;
    }
}

// ---------------------------------------------------------------------------
extern "C" void kernel_launch(void* const* d_in, const int* in_sizes, int n_in,
                              void* d_out, int out_size, void* d_ws, size_t ws_size,
                              hipStream_t stream) {
    const float* x     = (const float*)d_in[0];
    const int*   ei    = (const int*)  d_in[1];
    const float* ea    = (const float*)d_in[2];
    const float* mask  = (const float*)d_in[3];
    const int*   batch = (const int*)  d_in[4];
    const float* We    = (const float*)d_in[5];
    const float* be    = (const float*)d_in[6];
    const float* W1    = (const float*)d_in[7];
    const float* b1    = (const float*)d_in[8];
    const float* W2    = (const float*)d_in[9];
    const float* b2    = (const float*)d_in[10];
    const float* epsp  = (const float*)d_in[11];
    const float* gamma = (const float*)d_in[12];
    const float* beta  = (const float*)d_in[13];
    const float* Wv    = (const float*)d_in[14];
    const float* bv    = (const float*)d_in[15];
    float* out = (float*)d_out;

    const int N = in_sizes[0] / D_DIM;   // 50000
    const int E = in_sizes[1] / 2;       // 600000

    // Workspace: agg[N*128] | psum[512*128] | pcount[512] | pemb[512*128]
    float* ws     = (float*)d_ws;
    float* agg    = ws;
    float* psum   = agg + (long)N * D_DIM;
    float* pcount = psum + B_GRAPHS * K_PKT * D_DIM;
    float* pemb   = pcount + B_GRAPHS * K_PKT;
    const long zeroN = (long)N * D_DIM + B_GRAPHS * K_PKT * D_DIM + B_GRAPHS * K_PKT;

    zero_kernel<<<2048, 256, 0, stream>>>(ws, zeroN);

    const int nGroups = E / 16;                       // 37500
    edge_aggregate_kernel<<<(nGroups + 7) / 8, 256, 0, stream>>>(
        x, ei, ea, We, be, agg, E);

    const int nTiles = N / 16;                        // 3125
    node_mlp_kernel<<<(nTiles + 1) / 2, 64, 0, stream>>>(
        x, agg, W1, b1, W2, b2, epsp, out, N);

    const long nPairs = (long)N * K_PKT;              // 400000
    pocket_pool_kernel<<<(int)((nPairs + 7) / 8), 256, 0, stream>>>(
        out, mask, batch, psum, pcount, N);

    pocket_emb_kernel<<<16, 64, 0, stream>>>(psum, pcount, Wv, bv, pemb);

    finalize_kernel<<<(N + 7) / 8, 256, 0, stream>>>(
        out, pemb, mask, batch, gamma, beta, N);
}